// Pointer_32323923869835
// MI455X (gfx1250) — compile-verified
//
#include <hip/hip_runtime.h>
#include <math.h>

typedef __attribute__((ext_vector_type(2))) float v2f;
typedef __attribute__((ext_vector_type(4))) float v4f;
typedef __attribute__((ext_vector_type(8))) float v8f;

#define Bsz  4
#define DECn 256
#define ENCn 1024
#define Ddim 1024
#define Adim 144
#define Vsz  50257
#define NROWS (Bsz * DECn)                 // 1024
#define NV    ((size_t)NROWS * (size_t)Vsz) // 51,463,168 (divisible by 4)

// ---------------- Kernel 1: p_gen = sigmoid(hidden @ W_pgen + b) ----------------
__global__ __launch_bounds__(256) void pgen_kernel(const float* __restrict__ hidden,
                                                   const float* __restrict__ Wp,
                                                   const float* __restrict__ bp,
                                                   float* __restrict__ pgen_out) {
    __shared__ float red[256];
    const int n = blockIdx.x;       // row in [0, 1024)
    const int t = threadIdx.x;
    v4f h = *(const v4f*)(hidden + (size_t)n * Ddim + (size_t)t * 4);
    v4f w = *(const v4f*)(Wp + (size_t)t * 4);
    red[t] = h.x * w.x + h.y * w.y + h.z * w.z + h.w * w.w;
    __syncthreads();
    for (int s = 128; s > 0; s >>= 1) {
        if (t < s) red[t] += red[t + s];
        __syncthreads();
    }
    if (t == 0) {
        float x = red[0] + bp[0];
        pgen_out[n] = 1.0f / (1.0f + expf(-x));
    }
}

// ---------------- Kernel 2: out = p_gen * probs (streaming, NT hints) ----------------
__global__ __launch_bounds__(256) void base_kernel(const float* __restrict__ probs,
                                                   const float* __restrict__ pgen,
                                                   float* __restrict__ out) {
    const unsigned i4 = blockIdx.x * 256u + threadIdx.x;   // float4 index, exact grid
    const unsigned i  = i4 * 4u;
    const unsigned n0 = i / Vsz;          // constant divisor -> mulhi
    const unsigned n3 = (i + 3u) / Vsz;
    if (n0 == n3) {
        const float pg = pgen[n0];
        v4f p = __builtin_nontemporal_load(((const v4f*)probs) + (size_t)i4);
        v4f r = p * pg;
        __builtin_nontemporal_store(r, ((v4f*)out) + (size_t)i4);
    } else {  // float4 straddles a row boundary (V is odd) — rare scalar path
        for (int k = 0; k < 4; ++k) {
            const unsigned j = i + (unsigned)k;
            const unsigned n = j / Vsz;
            out[j] = pgen[n] * probs[j];
        }
    }
}

// ---------------- Kernel 3: WMMA dot-144 + relu + scaled scatter-add ----------------
// One wave handles 16 encoder rows of one (b,d) via V_WMMA_F32_16X16X4_F32.
// A tile (16x4 f32): lane L -> M = L&15, K pair = (L>>4)*2  => one b64 load/lane/chunk.
// B tile broadcasts the W_add chunk into all 16 columns so every column of D holds
// the 16 dot products; lanes 0 and 16 extract, relu, scale by (1-p_gen), atomic scatter.
__global__ __launch_bounds__(256) void scatter_kernel(const float* __restrict__ attn,
                                                      const float* __restrict__ Wadd,
                                                      const float* __restrict__ badd,
                                                      const int*   __restrict__ ids,
                                                      const float* __restrict__ pgen,
                                                      float* __restrict__ out) {
    const int tid  = threadIdx.x;
    const int lane = tid & 31;
    const int wave = tid >> 5;
    const int gid  = blockIdx.x * 8 + wave;   // tile id, [0, 65536) exactly
    const int n    = gid >> 6;                // b*DEC + d
    const int e0   = (gid & 63) << 4;         // base encoder row of the 16-row tile
    const int m    = lane & 15;               // M row this lane supplies
    const int kb   = (lane >> 4) << 1;        // K sub-pair: 0 or 2

    const float* arow = attn + ((size_t)n * ENCn + (size_t)(e0 + m)) * Adim + kb;
    const float* wrow = Wadd + kb;

    v8f c = {};
    #pragma unroll
    for (int ck = 0; ck < Adim / 4; ++ck) {   // 36 chunks of K=4
        v2f a = *(const v2f*)(arow + ck * 4); // K = ck*4+kb, ck*4+kb+1 for row M
        v2f b = *(const v2f*)(wrow + ck * 4); // same K pair, broadcast across N
        c = __builtin_amdgcn_wmma_f32_16x16x4_f32(false, a, false, b,
                                                  (short)0, c, false, false);
    }

    if (m == 0) {                              // lanes 0 (M=0..7) and 16 (M=8..15)
        const int   mbase = (lane >> 4) << 3;
        const int   bIdx  = n >> 8;            // n / DEC
        const float scale = 1.0f - pgen[n];
        const float ba    = badd[0];
        float* orow = out + (size_t)n * Vsz;
        #pragma unroll
        for (int i = 0; i < 8; ++i) {
            float add = c[i] + ba;
            add = add > 0.0f ? add : 0.0f;     // relu
            const float val = scale * add;
            const int   tok = ids[bIdx * ENCn + e0 + mbase + i];
            atomicAdd(orow + tok, val);        // duplicate tokens + 64 waves/row share targets
        }
    }
}

extern "C" void kernel_launch(void* const* d_in, const int* in_sizes, int n_in,
                              void* d_out, int out_size, void* d_ws, size_t ws_size,
                              hipStream_t stream) {
    const int*   ids    = (const int*)d_in[0];
    const float* attn   = (const float*)d_in[1];
    const float* hidden = (const float*)d_in[2];
    const float* probs  = (const float*)d_in[3];
    const float* Wp     = (const float*)d_in[4];
    const float* bp     = (const float*)d_in[5];
    const float* Wa     = (const float*)d_in[6];
    const float* ba     = (const float*)d_in[7];

    float* out  = (float*)d_out;
    float* pg   = out + NV;                    // tuple output #2 lives after out

    pgen_kernel<<<NROWS, 256, 0, stream>>>(hidden, Wp, bp, pg);

    const unsigned grid2 = (unsigned)(NV / 4 / 256);        // 50257, exact
    base_kernel<<<grid2, 256, 0, stream>>>(probs, pg, out);

    const unsigned grid3 = (NROWS * (ENCn / 16)) / 8;       // 8192, exact (8 waves/block)
    scatter_kernel<<<grid3, 256, 0, stream>>>(attn, Wa, ba, ids, pg, out);
}